// Experts_15126874816803
// MI455X (gfx1250) — compile-verified
//
#include <hip/hip_runtime.h>
#include <hip/hip_bf16.h>
#include <math.h>
#include <stdint.h>

// MoE Experts kernel for MI455X (gfx1250).
// dispatch (atomicAdd slots) -> grouped GEMM1 + exact GELU -> grouped GEMM2.
// Math in full f32 via V_WMMA_F32_16X16X4_F32 (wave32 matrix pipe).
// A tile staged with GLOBAL_LOAD_ASYNC_TO_LDS_B128 (ASYNCcnt path);
// B tile stored k-paired in LDS so every WMMA B fragment is one ds_load_b64.

typedef float v2f __attribute__((ext_vector_type(2)));
typedef float v8f __attribute__((ext_vector_type(8)));
typedef int   v4i __attribute__((ext_vector_type(4)));
typedef v4i __attribute__((address_space(1)))* gptr_v4i;   // global (prints as __device__)
typedef v4i __attribute__((address_space(3)))* lptr_v4i;   // LDS

#define NUM_TOKENS 32768
#define DMODEL     512
#define FFN        2048
#define NEXPERTS   64
#define CAPACITY   1024

#if defined(__AMDGCN__) && __has_builtin(__builtin_amdgcn_global_load_async_to_lds_b128)
#define USE_ASYNC_LDS 1
#else
#define USE_ASYNC_LDS 0
#endif

__global__ void zero_counts_kernel(int* counts) {
    counts[threadIdx.x] = 0;
}

__global__ void build_dispatch_kernel(const int* __restrict__ order,
                                      int* __restrict__ counts,
                                      int* __restrict__ rows) {
    int t = blockIdx.x * blockDim.x + threadIdx.x;
    if (t >= NUM_TOKENS) return;
    int e = order[t];
    int slot = atomicAdd(&counts[e], 1);
    if (slot < CAPACITY) rows[e * CAPACITY + slot] = t;
}

__device__ __forceinline__ float gelu_exact(float x) {
    return 0.5f * x * (1.0f + erff(x * 0.70710678118654752440f));
}

// Grouped GEMM over one expert's gathered rows.
//   dst[t, n] = act( sum_k src[t, k] * W[e, k, n] + bias[e, n] )
// Block tile: 32 rows x 128 cols; 8 waves, each wave a 16x32 C tile.
// K staged through LDS in 32-deep chunks; inner product via f32 WMMA (K=4).
template <int KDIM, int NDIM, bool DO_GELU>
__global__ __launch_bounds__(256)
void expert_gemm_kernel(const float* __restrict__ src,
                        const float* __restrict__ W,
                        const float* __restrict__ bias,
                        float* __restrict__ dst,
                        const int* __restrict__ rows,
                        const int* __restrict__ counts)
{
    constexpr int KC   = 32;   // K chunk depth
    constexpr int LDA  = 36;   // A LDS stride (floats): 144B rows -> 16B aligned, bank-spread
    constexpr int LDB2 = 144;  // B LDS stride in float2 units: 1152B rows; k-halves 32 banks apart
    __shared__ float As[32 * LDA];
    __shared__ float BsP[(KC / 2) * LDB2 * 2];  // k-paired: float2 {W[2p][n], W[2p+1][n]} at (p, n)
    __shared__ int   tok[32];

    const int e = blockIdx.z;
    int cnt = counts[e];
    if (cnt > CAPACITY) cnt = CAPACITY;
    const int row0 = blockIdx.y * 32;
    if (row0 >= cnt) return;                     // block-uniform exit
    const int n0 = blockIdx.x * 128;

    const int tid  = threadIdx.x;
    const int lane = tid & 31;
    const int wave = tid >> 5;
    const int wm   = wave >> 2;                  // 0..1 : row half of block tile
    const int wn   = wave & 3;                   // 0..3 : 32-col slice

    if (tid < 32) {
        int r = row0 + tid;
        tok[tid] = (r < cnt) ? rows[e * CAPACITY + r] : -1;
    }
    __syncthreads();

    const float* Wb = W + (size_t)e * KDIM * NDIM + n0;

    v8f c0 = {};                                 // cols [wn*32, wn*32+16)
    v8f c1 = {};                                 // cols [wn*32+16, wn*32+32)

    // A global->LDS mapping: one 16B copy per thread per chunk (chunk-invariant slot)
    const int arow = tid >> 3;                   // 0..31
    const int ak4  = (tid & 7) * 4;              // 0..28
    const int atok = tok[arow];
    float* a_slot = &As[arow * LDA + ak4];
    if (atok < 0) {                              // padded rows stay zero for all chunks
        float4 z = {0.f, 0.f, 0.f, 0.f};
        *(float4*)a_slot = z;
    }

    // WMMA fragment lane mapping (ISA 7.12.2)
    const int m     = lane & 15;                 // A row / B col within fragment
    const int khalf = lane >> 4;                 // 0: K=+0,+1  1: K=+2,+3

    for (int k0 = 0; k0 < KDIM; k0 += KC) {
        // ---- stage A (gathered token rows) ----
        if (atok >= 0) {
            const float* g = src + (size_t)atok * KDIM + k0 + ak4;
#if USE_ASYNC_LDS
            __builtin_amdgcn_global_load_async_to_lds_b128(
                (gptr_v4i)(uintptr_t)g,
                (lptr_v4i)(uint32_t)(uintptr_t)a_slot,
                0, 0);
#else
            *(float4*)a_slot = *(const float4*)g;
#endif
        }

        // ---- stage B (weights) in k-paired layout: 4x (2 b64 loads -> 1 b128 store) ----
#pragma unroll
        for (int i = 0; i < 4; ++i) {
            int lin = tid + i * 256;             // 0..1023
            int p   = lin >> 6;                  // 0..15 : k-pair row
            int n   = (lin & 63) * 2;            // 0..126: even column
            const float* g = Wb + (size_t)(k0 + 2 * p) * NDIM + n;
            float2 r0 = *(const float2*)(g);
            float2 r1 = *(const float2*)(g + NDIM);
            float4 v; v.x = r0.x; v.y = r1.x; v.z = r0.y; v.w = r1.y;
            *(float4*)&BsP[(p * LDB2 + n) * 2] = v;
        }
        // prefetch next weight chunk toward L2 (global_prefetch_b8)
        if (k0 + KC < KDIM)
            __builtin_prefetch(Wb + (size_t)(k0 + KC + (tid >> 5)) * NDIM + (tid & 31) * 4, 0, 1);

#if USE_ASYNC_LDS
        asm volatile("s_wait_asynccnt 0x0" ::: "memory");
#endif
        __syncthreads();

        // ---- 8 K-steps of 4 per chunk; wave-uniform control flow ----
#pragma unroll
        for (int kk = 0; kk < KC; kk += 4) {
            // A frag: lane holds (m, k0+kk+2*khalf +{0,1}) -> one aligned b64
            v2f a = *(const v2f*)&As[(wm * 16 + m) * LDA + kk + khalf * 2];
            // B frags: pair row (kk/2 + khalf) -> vgpr0 = K even, vgpr1 = K odd; one b64 each
            const int prow = (kk >> 1) + khalf;
            v2f b0  = *(const v2f*)&BsP[(prow * LDB2 + wn * 32 + m) * 2];
            v2f b1v = *(const v2f*)&BsP[(prow * LDB2 + wn * 32 + 16 + m) * 2];
            c0 = __builtin_amdgcn_wmma_f32_16x16x4_f32(false, a, false, b0,
                                                       (short)0, c0, false, false);
            c1 = __builtin_amdgcn_wmma_f32_16x16x4_f32(false, a, false, b1v,
                                                       (short)0, c1, false, false);
        }
        __syncthreads();
    }

    // ---- epilogue: C layout = vgpr r -> M = r + 8*(lane>=16), N = lane&15 ----
    const int ncol  = lane & 15;
    const int rbase = (lane >> 4) * 8;
#pragma unroll
    for (int r = 0; r < 8; ++r) {
        int mrow = wm * 16 + r + rbase;
        int t = tok[mrow];
        if (t < 0) continue;
        int gn0 = n0 + wn * 32 + ncol;
        int gn1 = gn0 + 16;
        float v0 = c0[r] + bias[e * NDIM + gn0];
        float v1 = c1[r] + bias[e * NDIM + gn1];
        if (DO_GELU) { v0 = gelu_exact(v0); v1 = gelu_exact(v1); }
        dst[(size_t)t * NDIM + gn0] = v0;
        dst[(size_t)t * NDIM + gn1] = v1;
    }
}

extern "C" void kernel_launch(void* const* d_in, const int* in_sizes, int n_in,
                              void* d_out, int out_size, void* d_ws, size_t ws_size,
                              hipStream_t stream) {
    (void)in_sizes; (void)n_in; (void)out_size; (void)ws_size;

    const float* x     = (const float*)d_in[0];   // [B,S,D] f32
    const int*   order = (const int*)  d_in[1];   // [T] i32
    const float* w1    = (const float*)d_in[2];   // [E,D,F] f32
    const float* b1    = (const float*)d_in[3];   // [E,F]   f32
    const float* w2    = (const float*)d_in[4];   // [E,F,D] f32
    const float* b2    = (const float*)d_in[5];   // [E,D]   f32
    float* out = (float*)d_out;                   // [B,S,D] f32

    // Workspace layout: counts(256B) | rows(E*CAP i32 = 256KB) | h(T*F f32 = 256MB)
    char* ws    = (char*)d_ws;
    int*  counts = (int*)ws;
    int*  rows   = (int*)(ws + 256);
    float* h     = (float*)(ws + 256 + (size_t)NEXPERTS * CAPACITY * sizeof(int));

    zero_counts_kernel<<<1, NEXPERTS, 0, stream>>>(counts);
    build_dispatch_kernel<<<NUM_TOKENS / 256, 256, 0, stream>>>(order, counts, rows);

    // GEMM1: [rows_e x 512] @ [512 x 2048] + b1, exact GELU -> h[t, f]
    expert_gemm_kernel<DMODEL, FFN, true>
        <<<dim3(FFN / 128, CAPACITY / 32, NEXPERTS), 256, 0, stream>>>(
            x, w1, b1, h, rows, counts);

    // GEMM2: [rows_e x 2048] @ [2048 x 512] + b2 -> out[t, d]
    expert_gemm_kernel<FFN, DMODEL, false>
        <<<dim3(DMODEL / 128, CAPACITY / 32, NEXPERTS), 256, 0, stream>>>(
            h, w2, b2, out, rows, counts);
}